// LIFNeuron_3642132267479
// MI455X (gfx1250) — compile-verified
//
#include <hip/hip_runtime.h>
#include <cstdint>

// ---------------- LIF neuron scan, CDNA5 (gfx1250) ----------------
// N neurons (one lane each), T timesteps (serial recurrence).
// Chunk input tiles (32 rows x 32 timesteps, row stride T) are DMA'd into a
// padded LDS ring by the Tensor Data Mover (tensor_load_to_lds, TENSORcnt),
// 6 chunks in flight. TDM hardware padding (4 DWORDs per 32-DWORD row)
// reproduces the 144B LDS row stride the compute lanes read. Outputs are
// staged in an LDS tile and written as full-cacheline global_store_b128.
// Fallback path: global_load_async_to_lds_b128 (ASYNCcnt).

#define WAVE    32
#define CHUNK   32               // timesteps per chunk
#define NBUF    8                // LDS ring buffers per wave
#define DEPTH   6                // chunks in flight
#define ROWF    (CHUNK + 4)      // padded row stride (36 floats = 144B)
#define T_FIXED 8192
#define LPC     (CHUNK / 4)      // b128 ops per chunk per lane (8)

#define WAIT_DS0() asm volatile("s_wait_dscnt 0" ::: "memory")

#if __has_builtin(__builtin_amdgcn_tensor_load_to_lds)
#define USE_TDM 1
#else
#define USE_TDM 0
#endif

#if USE_TDM
typedef unsigned int u32x4 __attribute__((ext_vector_type(4)));
typedef int          i32x4 __attribute__((ext_vector_type(4)));
typedef int          i32x8 __attribute__((ext_vector_type(8)));

// One TDM descriptor moves the whole 32x32-element tile (rows strided by T)
// into LDS with hardware row padding (128B data + 16B pad = 144B stride).
__device__ __forceinline__ void issue_chunk(const float* I, int T, int n0,
                                            unsigned lds_byte, int c,
                                            int, int) {
    unsigned long long ga = (unsigned long long)(uintptr_t)
        (I + (size_t)n0 * (size_t)T + (size_t)c * CHUNK);
    u32x4 g0;
    g0[0] = 1u;                                      // count=1 (valid, user)
    g0[1] = lds_byte;                                // lds_addr (bytes)
    g0[2] = (unsigned)(ga & 0xFFFFFFFFull);          // global_addr[31:0]
    g0[3] = (unsigned)((ga >> 32) & 0x1FFFFFFull)    // global_addr[56:32]
          | (2u << 30);                              // type=2 ("image")
    i32x8 g1;
    g1[0] = (2 << 16)        // data_size = 4B
          | (1 << 20)        // pad_enable
          | (4 << 22)        // pad_interval: 2^(4+1)=32 DWORDs (one row)
          | (3 << 25);       // pad_amount: 4 DWORDs (16B)
    g1[1] = (CHUNK & 0xFFFF) << 16;   // tensor_dim0 = 32 (low 16 at [31:16])
    g1[2] = (CHUNK & 0xFFFF) << 16;   // dim0 hi=0 | tensor_dim1 = 32 (lo16)
    g1[3] = (CHUNK & 0xFFFF) << 16;   // dim1 hi=0 | tile_dim0 = 32
    g1[4] = WAVE;                     // tile_dim1 = 32, tile_dim2 = 0
    g1[5] = T;                        // tensor_dim0_stride[31:0] (elements)
    g1[6] = 0;                        // stride0 hi | dim1_stride lo (2D: 0)
    g1[7] = 0;
    i32x4 gz = {0, 0, 0, 0};          // groups 2/3 unused (2D tile)
#if __has_include(<hip/amd_detail/amd_gfx1250_TDM.h>)
    i32x8 gz8 = {0, 0, 0, 0, 0, 0, 0, 0};
    __builtin_amdgcn_tensor_load_to_lds(g0, g1, gz, gz, gz8, 0);
#else
    __builtin_amdgcn_tensor_load_to_lds(g0, g1, gz, gz, 0);
#endif
}

// Tensor ops complete in order per wave: waiting TENSORcnt <= (#issued after
// chunk c) guarantees chunk c is resident.
__device__ __forceinline__ void wait_chunk(int later) {
    if      (later >= DEPTH - 1) __builtin_amdgcn_s_wait_tensorcnt(DEPTH - 1);
    else if (later == 4)         __builtin_amdgcn_s_wait_tensorcnt(4);
    else if (later == 3)         __builtin_amdgcn_s_wait_tensorcnt(3);
    else if (later == 2)         __builtin_amdgcn_s_wait_tensorcnt(2);
    else if (later == 1)         __builtin_amdgcn_s_wait_tensorcnt(1);
    else                         __builtin_amdgcn_s_wait_tensorcnt(0);
}

#else  // ---------- fallback: per-lane async loads ----------

__device__ __forceinline__ void async_load_b128(unsigned lds_byte_off,
                                                const float* gsrc) {
    unsigned long long ga = (unsigned long long)(uintptr_t)gsrc;
    asm volatile("global_load_async_to_lds_b128 %0, %1, off"
                 :: "v"(lds_byte_off), "v"(ga) : "memory");
}
#define WAIT_ASYNC(nn) asm volatile("s_wait_asynccnt %0" :: "n"(nn) : "memory")

__device__ __forceinline__ void issue_chunk(const float* I, int T, int n0,
                                            unsigned lds_byte, int c,
                                            int colg, int rsub) {
    #pragma unroll
    for (int g = 0; g < 8; ++g) {
        const int row = g * 4 + rsub;
        const float* src = I + (size_t)(n0 + row) * (size_t)T + c * CHUNK + colg;
        async_load_b128(lds_byte + (unsigned)(row * ROWF + colg) * 4u, src);
    }
}
__device__ __forceinline__ void wait_chunk(int later) {
    if      (later >= DEPTH - 1) WAIT_ASYNC(LPC * (DEPTH - 1));
    else if (later == 4)         WAIT_ASYNC(LPC * 4);
    else if (later == 3)         WAIT_ASYNC(LPC * 3);
    else if (later == 2)         WAIT_ASYNC(LPC * 2);
    else if (later == 1)         WAIT_ASYNC(LPC);
    else                         WAIT_ASYNC(0);
}
#endif

__global__ __launch_bounds__(WAVE)
void lif_scan_kernel(const float* __restrict__ I, float* __restrict__ out,
                     const float* tau_m_p, const float* v_th_p,
                     const float* v_reset_p, const float* r_m_p,
                     const float* v_rest_p, const float* refr_p,
                     int N, int T) {
    // [0 .. NBUF*32*ROWF)  : input ring buffers (TDM destination)
    // [NBUF*32*ROWF .. )   : output staging tile (32 x ROWF)
    __shared__ __align__(16) float lds[(NBUF + 1) * WAVE * ROWF];

    const int lane = threadIdx.x & (WAVE - 1);
    const int n0   = blockIdx.x * WAVE;
    const int n    = n0 + lane;
    if (n >= N) return;

    const int colg = (lane & 7) * 4;   // full-cacheline store remap
    const int rsub = lane >> 3;

    const float tau_m      = *tau_m_p;
    const float v_th       = *v_th_p;
    const float v_reset    = *v_reset_p;
    const float r_m        = *r_m_p;
    const float v_rest     = *v_rest_p;
    const float refractory = *refr_p;

    const float DTc      = 1.0f;
    const float a        = DTc / tau_m;
    const float b        = (r_m * 0.001f) * a;   // uA->nA fold + r_m/tau
    const float refr_set = refractory / DTc;

    const int nchunks = T / CHUNK;               // 256
    float* otile = &lds[NBUF * WAVE * ROWF];

    // ---- prime the DMA pipeline: chunks 0..DEPTH-1 ----
    #pragma unroll
    for (int p = 0; p < DEPTH; ++p)
        issue_chunk(I, T, n0,
                    (unsigned)(uintptr_t)&lds[(p % NBUF) * WAVE * ROWF],
                    p, colg, rsub);

    float v = v_rest, refr = 0.0f, Iprev = 0.0f;

    for (int c = 0; c < nchunks; ++c) {
        wait_chunk(nchunks - 1 - c);             // chunk c resident in LDS

        const float* row   = &lds[(c % NBUF) * WAVE * ROWF] + lane * ROWF;
        float*       myout = otile + lane * ROWF;

        #pragma unroll
        for (int s = 0; s < LPC; ++s) {
            float4 in4 = *(const float4*)(row + 4 * s);   // ds_load_b128
            float xs[4] = {in4.x, in4.y, in4.z, in4.w};
            float os[4];
            #pragma unroll
            for (int e = 0; e < 4; ++e) {
                const int t = c * CHUNK + s * 4 + e;
                float spike_f = 0.0f;
                if (t > 0) {   // folds away except one uniform check per chunk
                    float vint    = v + (v_rest - v) * a + Iprev * b;
                    bool  in_refr = refr > 0.0f;
                    bool  spike   = (!in_refr) && (vint >= v_th);
                    v    = (in_refr || spike) ? v_reset : vint;
                    refr = in_refr ? (refr - 1.0f) : (spike ? refr_set : refr);
                    spike_f = spike ? 1.0f : 0.0f;
                }
                os[e] = spike_f;
                Iprev = xs[e];      // out[t+1] consumes I[t]
            }
            float4 sp; sp.x = os[0]; sp.y = os[1]; sp.z = os[2]; sp.w = os[3];
            *(float4*)(myout + 4 * s) = sp;               // ds_store_b32x4
        }

        // Full-cacheline stores (DS ops in-order per wave -> tile reads see
        // this chunk's stores; compiler inserts the dscnt wait for the data).
        #pragma unroll
        for (int g = 0; g < 8; ++g) {
            const int row_o = g * 4 + rsub;
            float4 val = *(const float4*)(otile + row_o * ROWF + colg);
            *(float4*)(out + (size_t)(n0 + row_o) * (size_t)T + c * CHUNK + colg)
                = val;                                    // global_store_b128
        }

        // Refill ring: chunk c+DEPTH overwrites buffer (c-2)%NBUF whose DS
        // reads retired two iterations ago. Safety drain of DS queue first.
        const int pc = c + DEPTH;
        if (pc < nchunks) {
            WAIT_DS0();
            issue_chunk(I, T, n0,
                        (unsigned)(uintptr_t)&lds[(pc % NBUF) * WAVE * ROWF],
                        pc, colg, rsub);
        }
    }
}

extern "C" void kernel_launch(void* const* d_in, const int* in_sizes, int n_in,
                              void* d_out, int out_size, void* d_ws, size_t ws_size,
                              hipStream_t stream) {
    (void)n_in; (void)out_size; (void)d_ws; (void)ws_size;
    const float* I          = (const float*)d_in[0];
    const float* tau_m_p    = (const float*)d_in[1];
    const float* v_th_p     = (const float*)d_in[2];
    const float* v_reset_p  = (const float*)d_in[3];
    const float* r_m_p      = (const float*)d_in[4];
    const float* v_rest_p   = (const float*)d_in[5];
    const float* refr_p     = (const float*)d_in[6];
    float* out = (float*)d_out;

    const int T = T_FIXED;
    const int N = in_sizes[0] / T;              // 4096
    const int blocks = (N + WAVE - 1) / WAVE;   // 128 blocks, 1 wave each

    lif_scan_kernel<<<blocks, WAVE, 0, stream>>>(
        I, out, tau_m_p, v_th_p, v_reset_p, r_m_p, v_rest_p, refr_p, N, T);
}